// STConvBlock_28630251995692
// MI455X (gfx1250) — compile-verified
//
#include <hip/hip_runtime.h>
#include <hip/hip_bf16.h>
#include <cstdint>

#define BB 8
#define TT 64
#define NN 512
#define FF 32
#define GG 64
#define CC 64
#define EPSV 1e-5f

typedef float v2f __attribute__((ext_vector_type(2)));
typedef float v8f __attribute__((ext_vector_type(8)));

__device__ __forceinline__ v8f vzero8() {
    v8f z = {0.f, 0.f, 0.f, 0.f, 0.f, 0.f, 0.f, 0.f};
    return z;
}

// D = A(16x4) * B(4x16) + C, fp32 WMMA
__device__ __forceinline__ v8f wmma4(v2f a, v2f b, v8f c) {
    return __builtin_amdgcn_wmma_f32_16x16x4_f32(false, a, false, b, (short)0, c,
                                                 false, false);
}

// ---- CDNA5 async global->LDS copies (ASYNCcnt-tracked) ----
__device__ __forceinline__ void async_ld_b128(uint32_t lds_byte_off, const void* gptr) {
    asm volatile("global_load_async_to_lds_b128 %0, %1, off"
                 :
                 : "v"(lds_byte_off), "v"((uint64_t)(uintptr_t)gptr)
                 : "memory");
}
__device__ __forceinline__ void async_ld_b32(uint32_t lds_byte_off, const void* gptr) {
    asm volatile("global_load_async_to_lds_b32 %0, %1, off"
                 :
                 : "v"(lds_byte_off), "v"((uint64_t)(uintptr_t)gptr)
                 : "memory");
}
__device__ __forceinline__ void wait_async0() {
    asm volatile("s_wait_asynccnt 0" ::: "memory");
}

// ---------------- Kernel 0: inverse-sqrt degree ----------------
__global__ void __launch_bounds__(256) k_degree(const float* __restrict__ adj,
                                                float* __restrict__ dinv) {
    int wv = threadIdx.x >> 5, lane = threadIdx.x & 31;
    int row = blockIdx.x * 8 + wv;                 // b*N + i, 0..4095
    const float* r = adj + (size_t)row * NN;
    float s = (lane == 0) ? 1.0f : 0.0f;           // +1 from identity
    for (int j = lane; j < NN; j += 32) s += r[j];
    for (int off = 16; off > 0; off >>= 1) s += __shfl_xor(s, off, 32);
    if (lane == 0) dinv[row] = (s > 0.0f) ? rsqrtf(s) : 0.0f;
}

// ---------------- Kernel 1: materialize normalized adjacency ----------------
__global__ void __launch_bounds__(256) k_normadj(const float* __restrict__ adj,
                                                 const float* __restrict__ dinv,
                                                 float* __restrict__ nadj) {
    int idx = blockIdx.x * 256 + threadIdx.x;
    if (idx >= BB * NN * NN) return;
    int b = idx >> 18;
    int rem = idx & (NN * NN - 1);
    int i = rem >> 9, j = rem & (NN - 1);
    float v = adj[idx] + ((i == j) ? 1.0f : 0.0f);
    nadj[idx] = dinv[(b << 9) | i] * dinv[(b << 9) | j] * v;
}

// ---------------- Kernel 2: weight-norm + combine attention weights ----------------
__global__ void k_weights(const float* __restrict__ v1, const float* __restrict__ g1,
                          const float* __restrict__ v2, const float* __restrict__ g2,
                          const float* __restrict__ W1, const float* __restrict__ b1,
                          const float* __restrict__ W2, const float* __restrict__ b2,
                          float* __restrict__ wn1, float* __restrict__ wn2,
                          float* __restrict__ wc, float* __restrict__ bc) {
    int tid = threadIdx.x;  // 128 threads
    if (tid < 64) {
        int c = tid;
        float s = 0.f;
        for (int i = 0; i < 128; ++i) { float t = v1[c * 128 + i]; s += t * t; }
        float sc = g1[c] / sqrtf(s);
        for (int i = 0; i < 128; ++i) wn1[c * 128 + i] = v1[c * 128 + i] * sc;
        for (int j = 0; j < 64; ++j) wc[c * 64 + j] = W1[c * 64 + j] + W2[c * 64 + j];
        bc[c] = b1[c] + b2[c];
    } else if (tid < 128) {
        int c = tid - 64;
        float s = 0.f;
        for (int i = 0; i < 128; ++i) { float t = v2[c * 128 + i]; s += t * t; }
        float sc = g2[c] / sqrtf(s);
        for (int i = 0; i < 128; ++i) wn2[c * 128 + i] = v2[c * 128 + i] * sc;
    }
}

// ---------------- Kernel 3: GCN (WMMA): h = nadj @ x ; gcn = h @ gw + gb ----------------
// One workgroup per (pair of 16-row node tiles, batch, T-half). The two A tiles
// share every B operand: 4 WMMAs per 4 global loads.
__global__ void __launch_bounds__(256) k_gcn(const float* __restrict__ nadj,
                                             const float* __restrict__ x,
                                             const float* __restrict__ gw,
                                             const float* __restrict__ gb,
                                             float* __restrict__ gcn) {
    __shared__ float adjL[32 * 516];       // 32 adjacency rows, 16B-aligned stride
    __shared__ float gwL[32 * 64];         // [f][g]
    __shared__ float gbL[64];
    __shared__ float hL[8 * 16 * 33];      // per-wave h tile bounce buffer

    int tid = threadIdx.x;
    int ipair = blockIdx.x;                // 0..15
    int b = blockIdx.y;                    // 0..7
    int tz = blockIdx.z;                   // 0..1
    int i0 = ipair * 32;
    int wv = tid >> 5, lane = tid & 31;
    int m16 = lane & 15;
    int klo = (lane >> 4) << 1;            // 0 or 2 (K sub-rows per half-wave)
    int mrow = (lane >> 4) << 3;           // 0 or 8 (D-layout row base)

    // async-stage 32 adjacency rows: 128 chunks of 512B (b128 per lane)
    const float* arow = nadj + ((size_t)b * NN + i0) * NN;
    uint32_t adjBase = (uint32_t)(uintptr_t)&adjL[0];
    for (int c = wv; c < 128; c += 8) {
        int m = c >> 2, sub = c & 3;
        const float* gp = arow + (size_t)m * NN + sub * 128 + lane * 4;
        async_ld_b128(adjBase + (uint32_t)((m * 516 + sub * 128 + lane * 4) * 4), gp);
    }
    for (int idx = tid; idx < 32 * 64; idx += 256) gwL[idx] = gw[idx];
    if (tid < 64) gbL[tid] = gb[tid];
    wait_async0();
    __syncthreads();

    float* hb = hL + wv * 16 * 33;

    for (int t = tz * 32 + wv; t < tz * 32 + 32; t += 8) {
        const float* xb = x + (((size_t)b * TT + t) * NN) * FF;
        v8f acc00 = vzero8(), acc01 = vzero8();
        v8f acc10 = vzero8(), acc11 = vzero8();
        for (int j = 0; j < NN; j += 4) {
            v2f a0, a1;
            a0.x = adjL[m16 * 516 + j + klo];
            a0.y = adjL[m16 * 516 + j + klo + 1];
            a1.x = adjL[(16 + m16) * 516 + j + klo];
            a1.y = adjL[(16 + m16) * 516 + j + klo + 1];
            const float* xr0 = xb + (size_t)(j + klo) * FF;
            v2f bv0, bv1;
            bv0.x = xr0[m16];          bv0.y = xr0[FF + m16];
            bv1.x = xr0[16 + m16];     bv1.y = xr0[FF + 16 + m16];
            acc00 = wmma4(a0, bv0, acc00);
            acc01 = wmma4(a0, bv1, acc01);
            acc10 = wmma4(a1, bv0, acc10);
            acc11 = wmma4(a1, bv1, acc11);
        }
#pragma unroll
        for (int it2 = 0; it2 < 2; ++it2) {
            v8f h0 = it2 ? acc10 : acc00;
            v8f h1 = it2 ? acc11 : acc01;
            // D-layout -> A-layout via per-wave LDS
#pragma unroll
            for (int r = 0; r < 8; ++r) {
                hb[(mrow + r) * 33 + m16]      = h0[r];
                hb[(mrow + r) * 33 + 16 + m16] = h1[r];
            }
            // second matmul: [16x32] @ [32x64]
            v8f og[4] = {vzero8(), vzero8(), vzero8(), vzero8()};
            for (int kk = 0; kk < 8; ++kk) {
                int kb = kk * 4 + klo;
                v2f a2;
                a2.x = hb[m16 * 33 + kb];
                a2.y = hb[m16 * 33 + kb + 1];
#pragma unroll
                for (int gt = 0; gt < 4; ++gt) {
                    v2f bg;
                    bg.x = gwL[kb * 64 + gt * 16 + m16];
                    bg.y = gwL[(kb + 1) * 64 + gt * 16 + m16];
                    og[gt] = wmma4(a2, bg, og[gt]);
                }
            }
            float* outp = gcn + (((size_t)b * TT + t) * NN + i0 + it2 * 16) * GG;
#pragma unroll
            for (int gt = 0; gt < 4; ++gt) {
                int g = gt * 16 + m16;
                float bias = gbL[g];
#pragma unroll
                for (int r = 0; r < 8; ++r)
                    outp[(size_t)(mrow + r) * GG + g] = og[gt][r] + bias;
            }
        }
    }
}

// ---------------- Kernel 4: TCN (WMMA, both layers + shortcut fused) ----------------
__global__ void __launch_bounds__(256) k_tcn(const float* __restrict__ gcn,
                                             const float* __restrict__ wn1,
                                             const float* __restrict__ b1c,
                                             const float* __restrict__ wn2,
                                             const float* __restrict__ b2c,
                                             float* __restrict__ tcn) {
    __shared__ float seqL[64 * 65];   // [g][t]
    __shared__ float o1L[64 * 65];    // [c][t]
    __shared__ float bL[128];

    int tid = threadIdx.x;
    int bn = blockIdx.x;
    int b = bn >> 9, n = bn & 511;

    if (tid < 64) bL[tid] = b1c[tid];
    else if (tid < 128) bL[tid] = b2c[tid - 64];

    // async-stage seq tile (strided gather, b32 per lane)
    uint32_t seqBase = (uint32_t)(uintptr_t)&seqL[0];
    int gch = tid & 63, tq = tid >> 6;
    for (int u = 0; u < 16; ++u) {
        int t = tq * 16 + u;
        const float* gp = gcn + (((size_t)b * TT + t) * NN + n) * GG + gch;
        async_ld_b32(seqBase + (uint32_t)((gch * 65 + t) * 4), gp);
    }
    wait_async0();
    __syncthreads();

    int wv = tid >> 5, lane = tid & 31;
    int m16 = lane & 15;
    int mrow = (lane >> 4) << 3;
    int gp2 = (lane >> 4);                 // 0/1: which im2col pair within x4 step
    int klo = gp2 << 1;
    int ct = wv & 3;                       // output-channel tile
    int tb = (wv >> 2) << 1;               // two time tiles: tb, tb+1

    // ---- layer 1 ----
    v8f a1[2] = {vzero8(), vzero8()};
    for (int kk = 0; kk < 32; ++kk) {
        int kb = kk * 4 + klo;             // always even: row 2g (t-2 tap)
        int g = kk * 2 + gp2;
        v2f a = *reinterpret_cast<const v2f*>(wn1 + (ct * 16 + m16) * 128 + kb);
#pragma unroll
        for (int u = 0; u < 2; ++u) {
            int t = (tb + u) * 16 + m16;
            v2f bv;
            bv.x = 0.f;
            if (t >= 2) bv.x = seqL[g * 65 + t - 2];   // k=0 tap (dilation 2)
            bv.y = seqL[g * 65 + t];                   // k=1 tap
            a1[u] = wmma4(a, bv, a1[u]);
        }
    }
#pragma unroll
    for (int u = 0; u < 2; ++u) {
        int t = (tb + u) * 16 + m16;
#pragma unroll
        for (int r = 0; r < 8; ++r) {
            int c = ct * 16 + mrow + r;
            float v = a1[u][r] + bL[c];
            o1L[c * 65 + t] = (v > 0.f) ? v : 0.f;
        }
    }
    __syncthreads();

    // ---- layer 2 + shortcut ----
    v8f a2[2] = {vzero8(), vzero8()};
    for (int kk = 0; kk < 32; ++kk) {
        int kb = kk * 4 + klo;
        int g = kk * 2 + gp2;
        v2f a = *reinterpret_cast<const v2f*>(wn2 + (ct * 16 + m16) * 128 + kb);
#pragma unroll
        for (int u = 0; u < 2; ++u) {
            int t = (tb + u) * 16 + m16;
            v2f bv;
            bv.x = 0.f;
            if (t >= 2) bv.x = o1L[g * 65 + t - 2];
            bv.y = o1L[g * 65 + t];
            a2[u] = wmma4(a, bv, a2[u]);
        }
    }
    float* outp = tcn + (size_t)bn * CC * TT;   // layout [b,n,c,t]
#pragma unroll
    for (int u = 0; u < 2; ++u) {
        int t = (tb + u) * 16 + m16;
#pragma unroll
        for (int r = 0; r < 8; ++r) {
            int c = ct * 16 + mrow + r;
            float v = a2[u][r] + bL[64 + c];
            v = (v > 0.f) ? v : 0.f;
            float o = v + seqL[c * 65 + t];         // shortcut (G==C)
            outp[(size_t)c * TT + t] = (o > 0.f) ? o : 0.f;
        }
    }
}

// ---------------- Kernel 5: attention (WMMA) + softmax + weighted sum + LayerNorm ----------------
__global__ void __launch_bounds__(128) k_attn(const float* __restrict__ tcn,
                                              const float* __restrict__ wc,
                                              const float* __restrict__ bc,
                                              const float* __restrict__ Vw,
                                              const float* __restrict__ Vbp,
                                              const float* __restrict__ lng,
                                              const float* __restrict__ lnb,
                                              float* __restrict__ out) {
    __shared__ float tL[64 * 68];   // tcn [c][t], 16B-aligned stride
    __shared__ float wL[64 * 68];   // Wc  [d][c]
    __shared__ float vwL[64], bcL[64];
    __shared__ float scoreL[64], awL[64], attnL[64];

    int tid = threadIdx.x;
    int bn = blockIdx.x;
    int wv = tid >> 5, lane = tid & 31;

    // async-stage both 64x64 tiles: 32 chunks of 512B each (b128 per lane)
    const float* tp = tcn + (size_t)bn * 4096;
    uint32_t tBase = (uint32_t)(uintptr_t)&tL[0];
    uint32_t wBase = (uint32_t)(uintptr_t)&wL[0];
    for (int c = wv; c < 32; c += 4) {
        int rr = c * 2 + (lane >> 4);
        uint32_t loff = (uint32_t)(((rr * 68) + (lane & 15) * 4) * 4);
        const float* gt = tp + c * 128 + lane * 4;
        const float* gw2 = wc + c * 128 + lane * 4;
        async_ld_b128(tBase + loff, gt);
        async_ld_b128(wBase + loff, gw2);
    }
    if (tid < 64) { vwL[tid] = Vw[tid]; bcL[tid] = bc[tid]; }
    wait_async0();
    __syncthreads();
    float Vb = Vbp[0];

    int m16 = lane & 15;
    int klo = (lane >> 4) << 1;
    int mrow = (lane >> 4) << 3;
    int ttile = wv;                         // 4 waves x 16 timesteps

    float sacc = 0.f;
    for (int dt = 0; dt < 4; ++dt) {
        v8f acc = vzero8();
        for (int kk = 0; kk < 16; ++kk) {
            int kb = kk * 4 + klo;
            v2f a;
            a.x = wL[(dt * 16 + m16) * 68 + kb];
            a.y = wL[(dt * 16 + m16) * 68 + kb + 1];
            v2f bv;
            bv.x = tL[kb * 68 + ttile * 16 + m16];
            bv.y = tL[(kb + 1) * 68 + ttile * 16 + m16];
            acc = wmma4(a, bv, acc);
        }
#pragma unroll
        for (int r = 0; r < 8; ++r) {
            int d = dt * 16 + mrow + r;
            float pre = tanhf(acc[r] + bcL[d]);
            sacc += pre * vwL[d];
        }
    }
    sacc += __shfl_xor(sacc, 16, 32);       // combine d-halves (lanes l, l^16)
    scoreL[ttile * 16 + m16] = sacc + Vb;   // both halves write identical value
    __syncthreads();

    if (tid < 64) {
        float mx = scoreL[0];
        for (int t = 1; t < 64; ++t) mx = fmaxf(mx, scoreL[t]);
        float se = 0.f;
        for (int t = 0; t < 64; ++t) se += expf(scoreL[t] - mx);
        awL[tid] = expf(scoreL[tid] - mx) / se;
    }
    __syncthreads();
    if (tid < 64) {
        float a = 0.f;
        for (int t = 0; t < 64; ++t) a += tL[tid * 68 + t] * awL[t];
        attnL[tid] = a;
    }
    __syncthreads();
    if (tid < 64) {
        float mu = 0.f;
        for (int c = 0; c < 64; ++c) mu += attnL[c];
        mu *= (1.0f / 64.0f);
        float var = 0.f;
        for (int c = 0; c < 64; ++c) { float d = attnL[c] - mu; var += d * d; }
        var *= (1.0f / 64.0f);
        out[(size_t)bn * 64 + tid] =
            (attnL[tid] - mu) * rsqrtf(var + EPSV) * lng[tid] + lnb[tid];
    }
}

extern "C" void kernel_launch(void* const* d_in, const int* in_sizes, int n_in,
                              void* d_out, int out_size, void* d_ws, size_t ws_size,
                              hipStream_t stream) {
    (void)in_sizes; (void)n_in; (void)out_size; (void)ws_size;

    const float* x     = (const float*)d_in[0];
    const float* adj   = (const float*)d_in[1];
    const float* gcn_w = (const float*)d_in[2];
    const float* gcn_b = (const float*)d_in[3];
    const float* v1    = (const float*)d_in[4];
    const float* g1    = (const float*)d_in[5];
    const float* b1c   = (const float*)d_in[6];
    const float* v2    = (const float*)d_in[7];
    const float* g2    = (const float*)d_in[8];
    const float* b2c   = (const float*)d_in[9];
    const float* W1    = (const float*)d_in[10];
    const float* b1    = (const float*)d_in[11];
    const float* W2    = (const float*)d_in[12];
    const float* b2    = (const float*)d_in[13];
    const float* Vw    = (const float*)d_in[14];
    const float* Vb    = (const float*)d_in[15];
    const float* lng   = (const float*)d_in[16];
    const float* lnb   = (const float*)d_in[17];
    float* out = (float*)d_out;

    float* ws   = (float*)d_ws;
    float* dinv = ws;                         // 4096
    float* nadj = dinv + 4096;                // 2,097,152
    float* wn1  = nadj + 2097152;             // 8192
    float* wn2  = wn1 + 8192;                 // 8192
    float* wc   = wn2 + 8192;                 // 4096
    float* bc   = wc + 4096;                  // 64
    float* gcnb = bc + 64;                    // 16,777,216  [b,t,n,g]
    float* tcnb = gcnb + 16777216;            // 16,777,216  [b,n,c,t]

    k_degree<<<512, 256, 0, stream>>>(adj, dinv);
    k_normadj<<<(BB * NN * NN + 255) / 256, 256, 0, stream>>>(adj, dinv, nadj);
    k_weights<<<1, 128, 0, stream>>>(v1, g1, v2, g2, W1, b1, W2, b2, wn1, wn2, wc, bc);
    dim3 ggrid(16, 8, 2);
    k_gcn<<<ggrid, 256, 0, stream>>>(nadj, x, gcn_w, gcn_b, gcnb);
    k_tcn<<<BB * NN, 256, 0, stream>>>(gcnb, wn1, b1c, wn2, b2c, tcnb);
    k_attn<<<BB * NN, 128, 0, stream>>>(tcnb, wc, bc, Vw, Vb, lng, lnb, out);
}